// FourierWarpSingleImage_70437463654728
// MI455X (gfx1250) — compile-verified
//
#include <hip/hip_runtime.h>
#include <stdint.h>

// Problem constants (from reference): B,H,W,C = 2,256,256,64; out = (B,H,W,2C) f32.
namespace {
constexpr int kB = 2;
constexpr int kH = 256;
constexpr int kW = 256;
constexpr int kC = 64;
constexpr int kPix = 16;            // pixels per block along W
constexpr int kCG  = kC / 4;        // 16 channel groups of float4
constexpr float kTwoPi3 = 2.0943951023931953f;   // 2*pi/3
constexpr float kSqrt3  = 1.7320508075688772f;   // sqrt(3)
constexpr float kC2     = -0.41614683654714241f; // cos(2)
constexpr float kS2     =  0.90929742682568170f; // sin(2)
}

typedef float v4f __attribute__((ext_vector_type(4)));

__global__ __launch_bounds__(256) void fourier_warp_kernel(
    const float* __restrict__ image,
    const float* __restrict__ offx,
    const float* __restrict__ offy,
    float* __restrict__ out) {
  __shared__ float s_off[2 * kPix];      // [0..15] = offsetx, [16..31] = offsety
  __shared__ float s_wx[kPix][3];
  __shared__ float s_wy[kPix][3];
  __shared__ int   s_ix[kPix];
  __shared__ int   s_iy[kPix];

  const int tid = threadIdx.x;
  const int w0  = blockIdx.x * kPix;
  const int h   = blockIdx.y;
  const int b   = blockIdx.z;
  const int rowbase = (b * kH + h) * kW + w0;

  // ---- Phase 0: async global->LDS stage of the 32 offset values (wave 0) ----
  if (tid < 32) {
    const float* src = (tid < kPix) ? (offx + rowbase + tid)
                                    : (offy + rowbase + (tid - kPix));
    // Low 32 bits of a flat LDS pointer == byte offset from the wave's LDS base,
    // which is exactly what GLOBAL_LOAD_ASYNC_TO_LDS_B32's VDST expects.
    unsigned ldsaddr = (unsigned)(uintptr_t)(&s_off[tid]);
    asm volatile("global_load_async_to_lds_b32 %0, %1, off"
                 :: "v"(ldsaddr), "v"(src)
                 : "memory");
    asm volatile("s_wait_asynccnt 0x0" ::: "memory");
  }
  __syncthreads();

  // ---- Phase 1: per-pixel separable real weights (Axi==Ayi==0 analytically) ----
  if (tid < kPix) {
    const float ox = s_off[tid];
    const float oy = s_off[kPix + tid];
    const int ix0 = (int)ox;            // trunc toward zero == jnp.fix
    const int iy0 = (int)oy;
    const float fx = ox - (float)ix0;
    const float fy = oy - (float)iy0;
    float sx, cx, sy, cy;
    sincosf(kTwoPi3 * fx, &sx, &cx);
    sincosf(kTwoPi3 * fy, &sy, &cy);
    s_wx[tid][0] = (1.0f - cx - kSqrt3 * sx) * (1.0f / 3.0f);
    s_wx[tid][1] = (1.0f + 2.0f * cx)        * (1.0f / 3.0f);
    s_wx[tid][2] = (1.0f - cx + kSqrt3 * sx) * (1.0f / 3.0f);
    s_wy[tid][0] = (1.0f - cy - kSqrt3 * sy) * (1.0f / 3.0f);
    s_wy[tid][1] = (1.0f + 2.0f * cy)        * (1.0f / 3.0f);
    s_wy[tid][2] = (1.0f - cy + kSqrt3 * sy) * (1.0f / 3.0f);
    s_ix[tid] = ix0;
    s_iy[tid] = iy0;
  }
  __syncthreads();

  // ---- Phase 2: branchless clamped 3x3 gather + weighted sum ----
  const int cg = tid & (kCG - 1);       // channel group (4 floats)
  const int p  = tid >> 4;              // pixel within block
  const int w  = w0 + p;

  const float wxv[3] = { s_wx[p][0], s_wx[p][1], s_wx[p][2] };
  const float wyv[3] = { s_wy[p][0], s_wy[p][1], s_wy[p][2] };
  const int ix0 = s_ix[p];
  const int iy0 = s_iy[p];

  v4f acc = (v4f){0.0f, 0.0f, 0.0f, 0.0f};

#pragma unroll
  for (int l = 0; l < 3; ++l) {
    const int yy  = h + iy0 + (l - 1);
    const bool yok = (yy >= 0) && (yy < kH);
    const int yyc = yok ? yy : 0;
    const float* rowp = image + ((size_t)(b * kH + yyc) * kW) * kC + 4 * cg;
#pragma unroll
    for (int k = 0; k < 3; ++k) {
      const int xx  = w + ix0 + (k - 1);
      const bool xok = (xx >= 0) && (xx < kW);
      const int xxc = xok ? xx : 0;
      const float wgt = (yok && xok) ? (wxv[k] * wyv[l]) : 0.0f;
      const v4f v = *reinterpret_cast<const v4f*>(rowp + (size_t)xxc * kC);
      acc += wgt * v;   // contracts to v_fma under -ffp-contract
    }
  }

  // out[b,h,w, 0:64] = cos(2)*S ; out[b,h,w, 64:128] = -sin(2)*S
  float* obase = out + ((size_t)(b * kH + h) * kW + w) * (2 * kC) + 4 * cg;
  const v4f fr = kC2 * acc;
  const v4f fi = -kS2 * acc;
  __builtin_nontemporal_store(fr, reinterpret_cast<v4f*>(obase));
  __builtin_nontemporal_store(fi, reinterpret_cast<v4f*>(obase + kC));
}

extern "C" void kernel_launch(void* const* d_in, const int* in_sizes, int n_in,
                              void* d_out, int out_size, void* d_ws, size_t ws_size,
                              hipStream_t stream) {
  (void)in_sizes; (void)n_in; (void)d_ws; (void)ws_size; (void)out_size;
  const float* image = (const float*)d_in[0];
  const float* offx  = (const float*)d_in[1];
  const float* offy  = (const float*)d_in[2];
  float* out = (float*)d_out;

  dim3 grid(kW / kPix, kH, kB);   // (16, 256, 2)
  dim3 block(kPix * kCG);         // 256 threads = 8 wave32
  fourier_warp_kernel<<<grid, block, 0, stream>>>(image, offx, offy, out);
}